// Net_20761871908940
// MI455X (gfx1250) — compile-verified
//
#include <hip/hip_runtime.h>
#include <hip/hip_bf16.h>
#include <math.h>

// ---------------------------------------------------------------------------
// MI455X (gfx1250) implementation.
// Bandwidth-bound network: ~66MB of fp32 weights streamed once -> ~2.9us floor
// at 23.3 TB/s. Heavy matmuls are (19 x 1000) @ (1000 x 1000)^T done with
// V_WMMA_F32_16X16X4_F32 (exact fp32 path), weights streamed to LDS with
// double-buffered GLOBAL_LOAD_ASYNC_TO_LDS_B128 (ASYNCcnt-tracked DMA).
// ---------------------------------------------------------------------------

typedef float v2f __attribute__((ext_vector_type(2)));
typedef float v8f __attribute__((ext_vector_type(8)));

#define NNODE 19
#define GDIM  1000
#define RSTR  68            // LDS row stride in floats (bank-conflict-free: 4r+k)

// src_sl / dst_sl (55 edges incl. self loops) from the reference, compile-time.
__constant__ int c_src[55] = {
  0,1,2,3,4,5,6,7,8,9,10,11,12,13,14,15,16,17,18,
  1,2,3,4,5,6,7,8,9,10,11,12,13,14,15,16,17,
  0,1,2,3,4,5,6,7,8,9,10,11,12,13,14,15,16,17,18};
__constant__ int c_dst[55] = {
  1,2,3,4,5,6,7,8,9,10,11,12,13,14,15,16,17,18,17,
  0,1,2,3,4,5,6,7,8,9,10,11,12,13,14,15,16,
  0,1,2,3,4,5,6,7,8,9,10,11,12,13,14,15,16,17,18};

// ---------------------------------------------------------------------------
// Generic 5x5 stride-2 conv (tiny: <=120K outputs total)
__global__ void conv5x5s2(const float* __restrict__ in, const float* __restrict__ w,
                          const float* __restrict__ b, float* __restrict__ out,
                          int Cin, int H, int W, int Cout, int OH, int OW,
                          int pad, int relu) {
  int idx = blockIdx.x * blockDim.x + threadIdx.x;
  int total = Cout * OH * OW;
  if (idx >= total) return;
  int ow = idx % OW;
  int oh = (idx / OW) % OH;
  int co = idx / (OW * OH);
  float acc = b[co];
  for (int ci = 0; ci < Cin; ++ci) {
    const float* ip = in + (size_t)ci * H * W;
    const float* wp = w + ((size_t)co * Cin + ci) * 25;
#pragma unroll
    for (int ky = 0; ky < 5; ++ky) {
      int iy = oh * 2 + ky - pad;
      if (iy < 0 || iy >= H) continue;
#pragma unroll
      for (int kx = 0; kx < 5; ++kx) {
        int ix = ow * 2 + kx - pad;
        if (ix < 0 || ix >= W) continue;
        acc += ip[iy * W + ix] * wp[ky * 5 + kx];
      }
    }
  }
  out[idx] = relu ? fmaxf(acc, 0.f) : acc;
}

// ---------------------------------------------------------------------------
// GEMV: y = act(W @ x + b). One wave per row; 128B/iteration coalesced reads.
__global__ void gemv_kernel(const float* __restrict__ W, const float* __restrict__ x,
                            const float* __restrict__ b, float* __restrict__ y,
                            int M, int K, int relu) {
  int wave = threadIdx.x >> 5;
  int lane = threadIdx.x & 31;
  int row  = blockIdx.x * (blockDim.x >> 5) + wave;
  if (row >= M) return;
  const float* wr = W + (size_t)row * K;
  float acc = 0.f;
  for (int k = lane; k < K; k += 32) acc += wr[k] * x[k];
#pragma unroll
  for (int off = 16; off > 0; off >>= 1) acc += __shfl_xor(acc, off, 32);
  if (lane == 0) {
    float v = acc + b[row];
    y[row] = relu ? fmaxf(v, 0.f) : v;
  }
}

// ---------------------------------------------------------------------------
// Async weight streaming: load a 16-row x CS-col fp32 tile into LDS.
// CS/8 GLOBAL_LOAD_ASYNC_TO_LDS_B128 instructions; 16B/lane contiguous.
template <int CS>
__device__ __forceinline__ void issue_async(const float* __restrict__ Wt, int K,
                                            int n0, int N, int kstart,
                                            unsigned ldsbase, int lane) {
#pragma unroll
  for (int j = 0; j < CS / 8; ++j) {
    int flat = (j * 32 + lane) * 4;      // flat float index within the tile
    int row  = flat / CS;                // CS constexpr -> shifts
    int kofs = flat % CS;
    int gr   = n0 + row;
    gr = (gr < N) ? gr : (N - 1);        // clamp; masked at use
    unsigned laddr = ldsbase + (unsigned)((row * RSTR + kofs) * 4);
    unsigned long long gaddr =
        (unsigned long long)(uintptr_t)(Wt + (size_t)gr * K + (kstart + kofs));
    asm volatile("global_load_async_to_lds_b128 %0, %1, off"
                 :: "v"(laddr), "v"(gaddr) : "memory");
  }
}

// Consume one LDS-resident chunk: CS/4 steps of (ds_load_b64 B, 2x WMMA).
template <int CS>
__device__ __forceinline__ void process_chunk(const float* __restrict__ buf,
                                              const float* __restrict__ a0p,
                                              const float* __restrict__ a1p,
                                              int kbase, int r, int khalf,
                                              float bval, v8f& acc0, v8f& acc1) {
#pragma unroll
  for (int kk = 0; kk < CS; kk += 4) {
    const int kkl = kk + khalf * 2;      // lanes 0-15: K+0/1 ; 16-31: K+2/3
    v2f bb = *(const v2f*)(buf + r * RSTR + kkl);
    v2f a0 = *(const v2f*)(a0p + kbase + kkl);
    v2f a1 = *(const v2f*)(a1p + kbase + kkl);   // rows 16..31 zero-padded
    bb = bb * bval;                      // zero cols >= N (last tile only)
    acc0 = __builtin_amdgcn_wmma_f32_16x16x4_f32(false, a0, false, bb,
                                                 (short)0, acc0, false, false);
    acc1 = __builtin_amdgcn_wmma_f32_16x16x4_f32(false, a1, false, bb,
                                                 (short)0, acc1, false, false);
  }
}

// ---------------------------------------------------------------------------
// WMMA skinny GEMM: Y(19 x N) += A(32-row zero-padded x K) @ Wt(N x K)^T.
// One wave per 16-col N tile; K split 5x over blockIdx.y (200 each = 3x64+8);
// partials merged with hardware f32 atomics.
__global__ void __launch_bounds__(32)
gemm19_wmma(const float* __restrict__ A, const float* __restrict__ Wt,
            float* __restrict__ Y, int K, int N) {
  __shared__ float smem[2][16 * RSTR];
  const int lane  = threadIdx.x;         // wave32
  const int n0    = blockIdx.x * 16;
  const int kbeg  = blockIdx.y * 200;
  const int r     = lane & 15;
  const int khalf = lane >> 4;
  const float bval = (n0 + r < N) ? 1.f : 0.f;
  const float* a0p = A + (size_t)r * K;
  const float* a1p = A + (size_t)(16 + r) * K;   // zero-padded rows
  const unsigned lds0 = (unsigned)(uintptr_t)&smem[0][0]; // LDS offset = addr[31:0]
  const unsigned lds1 = (unsigned)(uintptr_t)&smem[1][0];

  v8f acc0 = {};
  v8f acc1 = {};

  // Double-buffered DMA pipeline over chunks {64,64,64,8}.
  issue_async<64>(Wt, K, n0, N, kbeg,       lds0, lane);
  issue_async<64>(Wt, K, n0, N, kbeg + 64,  lds1, lane);
  asm volatile("s_wait_asynccnt 0x8" ::: "memory");       // chunk0 landed
  process_chunk<64>(smem[0], a0p, a1p, kbeg, r, khalf, bval, acc0, acc1);
  asm volatile("s_wait_dscnt 0x0" ::: "memory");          // lds0 reads captured
  issue_async<64>(Wt, K, n0, N, kbeg + 128, lds0, lane);
  asm volatile("s_wait_asynccnt 0x8" ::: "memory");       // chunk1 landed
  process_chunk<64>(smem[1], a0p, a1p, kbeg + 64, r, khalf, bval, acc0, acc1);
  asm volatile("s_wait_dscnt 0x0" ::: "memory");          // lds1 reads captured
  issue_async<8>(Wt, K, n0, N, kbeg + 192, lds1, lane);
  asm volatile("s_wait_asynccnt 0x1" ::: "memory");       // chunk2 landed
  process_chunk<64>(smem[0], a0p, a1p, kbeg + 128, r, khalf, bval, acc0, acc1);
  asm volatile("s_wait_asynccnt 0x0" ::: "memory");       // tail landed
  process_chunk<8>(smem[1], a0p, a1p, kbeg + 192, r, khalf, bval, acc0, acc1);

  // C/D layout: VGPR i -> row i (+8 for lanes 16-31), col = n0 + (lane&15).
  const int col = n0 + r;
  if (col < N) {
#pragma unroll
    for (int i = 0; i < 8; ++i) {
      int row0 = i + khalf * 8;
      unsafeAtomicAdd(&Y[(size_t)row0 * N + col], acc0[i]);
    }
    if (khalf == 0) {                    // tile1 live rows: 16,17,18
#pragma unroll
      for (int i = 0; i < 3; ++i)
        unsafeAtomicAdd(&Y[(size_t)(16 + i) * N + col], acc1[i]);
    }
  }
}

// ---------------------------------------------------------------------------
// Small helper kernels -------------------------------------------------------
__global__ void zero_k(float* x, int n) {
  int i = blockIdx.x * blockDim.x + threadIdx.x;
  if (i < n) x[i] = 0.f;
}
__global__ void relu_k(float* x, int n) {
  int i = blockIdx.x * blockDim.x + threadIdx.x;
  if (i < n) x[i] = fmaxf(x[i], 0.f);
}
__global__ void set_rows_bias(float* Y, const float* b, int n) {  // 19 x GDIM
  int i = blockIdx.x * blockDim.x + threadIdx.x;
  if (i < n) Y[i] = b[i % GDIM];
}

// agg[n] = sum of neighbors (structure compile-time: {n-1}, {n+1 if n<=16}, {18->17})
__global__ void sage_agg(const float* __restrict__ X, float* __restrict__ AGG) {
  int idx = blockIdx.x * blockDim.x + threadIdx.x;
  if (idx >= NNODE * GDIM) return;
  int n = idx / GDIM, c = idx % GDIM;
  float acc = 0.f;
  if (n >= 1)  acc += X[(n - 1) * GDIM + c];
  if (n <= 16) acc += X[(n + 1) * GDIM + c];
  if (n == 17) acc += X[18 * GDIM + c];
  AGG[idx] = acc;
}

// First SAGE layer: 1-dim input, pure outer products.
__global__ void s1l1_kernel(const float* __restrict__ xg, const float* __restrict__ wl,
                            const float* __restrict__ bl, const float* __restrict__ wr,
                            float* __restrict__ Y) {
  int idx = blockIdx.x * blockDim.x + threadIdx.x;
  if (idx >= NNODE * GDIM) return;
  int n = idx / GDIM, m = idx % GDIM;
  float aggn = 0.f;
  if (n >= 1)  aggn += xg[n - 1];
  if (n <= 16) aggn += xg[n + 1];
  if (n == 17) aggn += xg[18];
  float v = aggn * wl[m] + bl[m] + xg[n] * wr[m];
  Y[idx] = fmaxf(v, 0.f);
}

// hs[n] = <H[n,:], a_s>, hd[n] = <H[n,:], a_d>; 19 blocks x 2 waves.
__global__ void gat_scores(const float* __restrict__ H, const float* __restrict__ as_,
                           const float* __restrict__ ad_, float* __restrict__ hs,
                           float* __restrict__ hd) {
  int n = blockIdx.x;
  int wave = threadIdx.x >> 5, lane = threadIdx.x & 31;
  const float* a = wave ? ad_ : as_;
  float acc = 0.f;
  for (int k = lane; k < GDIM; k += 32) acc += H[n * GDIM + k] * a[k];
#pragma unroll
  for (int off = 16; off > 0; off >>= 1) acc += __shfl_xor(acc, off, 32);
  if (lane == 0) (wave ? hd : hs)[n] = acc;
}

// Segment softmax over dst_sl (55 edges, 19 segments). One tiny block.
__global__ void gat_softmax(const float* __restrict__ hs, const float* __restrict__ hd,
                            float* __restrict__ alpha) {
  __shared__ float e[55], mx[NNODE], sm[NNODE];
  int t = threadIdx.x;
  if (t < 55) {
    float v = hs[c_src[t]] + hd[c_dst[t]];
    e[t] = v > 0.f ? v : 0.2f * v;   // leaky_relu(0.2)
  }
  __syncthreads();
  if (t < NNODE) {
    float m = -1e30f;
    for (int j = 0; j < 55; ++j) if (c_dst[j] == t) m = fmaxf(m, e[j]);
    float s = 0.f;
    for (int j = 0; j < 55; ++j) if (c_dst[j] == t) s += __expf(e[j] - m);
    mx[t] = m; sm[t] = s;
  }
  __syncthreads();
  if (t < 55) alpha[t] = __expf(e[t] - mx[c_dst[t]]) / sm[c_dst[t]];
}

// out[n,c] = relu( b[c] + sum_in alpha[e] * H[src(e), c] ); edge lists inlined.
__global__ void gat_agg(const float* __restrict__ H, const float* __restrict__ alpha,
                        const float* __restrict__ b, float* __restrict__ Y) {
  int idx = blockIdx.x * blockDim.x + threadIdx.x;
  if (idx >= NNODE * GDIM) return;
  int n = idx / GDIM, c = idx % GDIM;
  float acc = b[c];
  if (n >= 1)  acc += alpha[n - 1]  * H[(n - 1) * GDIM + c];   // edge n-1 -> n
  if (n <= 16) acc += alpha[19 + n] * H[(n + 1) * GDIM + c];   // edge n+1 -> n
  if (n == 17) acc += alpha[18]     * H[18 * GDIM + c];        // edge 18 -> 17
  acc += alpha[36 + n] * H[n * GDIM + c];                      // self loop
  Y[idx] = fmaxf(acc, 0.f);
}

// GCN normalized aggregation; degrees compile-time (3 at ends, 4 inside).
__device__ __forceinline__ float dinv19(int m) {
  return (m == 0 || m == 18) ? 0.57735026919f : 0.5f;          // 1/sqrt(deg)
}
__global__ void gcn_agg(const float* __restrict__ H, const float* __restrict__ b,
                        float* __restrict__ Y) {
  int idx = blockIdx.x * blockDim.x + threadIdx.x;
  if (idx >= NNODE * GDIM) return;
  int n = idx / GDIM, c = idx % GDIM;
  float dn = dinv19(n);
  float acc = b[c];
  if (n >= 1)  acc += dinv19(n - 1) * dn * H[(n - 1) * GDIM + c];
  if (n <= 16) acc += dinv19(n + 1) * dn * H[(n + 1) * GDIM + c];
  if (n == 17) acc += dinv19(18) * dn * H[18 * GDIM + c];
  acc += 2.f * dn * dn * H[n * GDIM + c];                      // self loop, ew=2
  Y[idx] = acc;
}

__global__ void mean19(const float* __restrict__ X, float* __restrict__ out) {
  int c = blockIdx.x * blockDim.x + threadIdx.x;
  if (c >= GDIM) return;
  float s = 0.f;
  for (int n = 0; n < NNODE; ++n) s += X[n * GDIM + c];
  out[c] = s * (1.f / 19.f);
}

// Two scalar heads: out[0]=value, out[1]=advantage. One block, 2 waves.
__global__ void heads_kernel(const float* __restrict__ f, const float* __restrict__ wv,
                             const float* __restrict__ bv, const float* __restrict__ wa,
                             const float* __restrict__ ba, float* __restrict__ out) {
  int lane = threadIdx.x & 31, h = threadIdx.x >> 5;
  const float* w = h ? wa : wv;
  float acc = 0.f;
  for (int k = lane; k < 256; k += 32) acc += w[k] * f[k];
#pragma unroll
  for (int off = 16; off > 0; off >>= 1) acc += __shfl_xor(acc, off, 32);
  if (lane == 0) out[h] = acc + (h ? ba[0] : bv[0]);
}

// ---------------------------------------------------------------------------
extern "C" void kernel_launch(void* const* d_in, const int* in_sizes, int n_in,
                              void* d_out, int out_size, void* d_ws, size_t ws_size,
                              hipStream_t stream) {
  const float* x      = (const float*)d_in[0];
  const float* state  = (const float*)d_in[1];
  const float* xg     = (const float*)d_in[2];
  const float* conv1_w = (const float*)d_in[3];
  const float* conv1_b = (const float*)d_in[4];
  const float* conv2_w = (const float*)d_in[5];
  const float* conv2_b = (const float*)d_in[6];
  const float* conv3_w = (const float*)d_in[7];
  const float* conv3_b = (const float*)d_in[8];
  const float* fc0_w = (const float*)d_in[9];
  const float* fc0_b = (const float*)d_in[10];
  const float* fc1_w = (const float*)d_in[11];
  const float* fc1_b = (const float*)d_in[12];
  const float* fc2_w = (const float*)d_in[13];
  const float* fc2_b = (const float*)d_in[14];
  const float* fc3_w = (const float*)d_in[15];
  const float* fc3_b = (const float*)d_in[16];
  const float* s1l1_wl = (const float*)d_in[17];
  const float* s1l1_bl = (const float*)d_in[18];
  const float* s1l1_wr = (const float*)d_in[19];
  const float* s1l2_wl = (const float*)d_in[20];
  const float* s1l2_bl = (const float*)d_in[21];
  const float* s1l2_wr = (const float*)d_in[22];
  const float* gat2_w  = (const float*)d_in[23];
  const float* gat2_as = (const float*)d_in[24];
  const float* gat2_ad = (const float*)d_in[25];
  const float* gat2_b  = (const float*)d_in[26];
  const float* s3l1_wl = (const float*)d_in[27];
  const float* s3l1_bl = (const float*)d_in[28];
  const float* s3l1_wr = (const float*)d_in[29];
  const float* s3l2_wl = (const float*)d_in[30];
  const float* s3l2_bl = (const float*)d_in[31];
  const float* s3l2_wr = (const float*)d_in[32];
  const float* gat4_w  = (const float*)d_in[33];
  const float* gat4_as = (const float*)d_in[34];
  const float* gat4_ad = (const float*)d_in[35];
  const float* gat4_b  = (const float*)d_in[36];
  const float* gcn5_w  = (const float*)d_in[37];
  const float* gcn5_b  = (const float*)d_in[38];
  const float* fc_graph_w = (const float*)d_in[39];
  const float* fc_graph_b = (const float*)d_in[40];
  const float* fc_fusion_w = (const float*)d_in[41];
  const float* fc_fusion_b = (const float*)d_in[42];
  const float* fc_value_w = (const float*)d_in[43];
  const float* fc_value_b = (const float*)d_in[44];
  const float* fc_adv_w   = (const float*)d_in[45];
  const float* fc_adv_b   = (const float*)d_in[46];

  float* ws   = (float*)d_ws;
  float* c1   = ws;              // 32*61*61 = 119072
  float* c2   = c1 + 119072;     // 32*29*29 = 26912
  float* c3   = c2 + 26912;      // 6272 (== flattened h)
  float* fc0o = c3 + 6272;       // 1024
  float* f    = fc0o + 1024;     // 420 fused: [img 256 | s 64 | g 100]
  float* s0   = f + 420;         // 64
  float* fo   = s0 + 64;         // 256 (fusion out)
  float* GA   = fo + 256;        // 32x1000 (rows 19..31 zero)
  float* GB   = GA + 32 * GDIM;  // 32x1000 (rows 19..31 zero)
  float* AGG  = GB + 32 * GDIM;  // 32x1000 (rows 19..31 zero)
  float* H    = AGG + 32 * GDIM; // 19x1000 (gemm output only)
  float* gm   = H + NNODE * GDIM;// 1000
  float* hs   = gm + GDIM;       // 19
  float* hd   = hs + NNODE;      // 19
  float* alp  = hd + NNODE;      // 55

  const int GN = NNODE * GDIM;          // 19000
  const int PN = (32 - NNODE) * GDIM;   // 13000 zero-pad rows
  const int EB = (GN + 255) / 256;
  const int PB = (PN + 255) / 256;
  const dim3 GG((GDIM + 15) / 16, 5);   // 63 N-tiles x 5 K-splits (200 each)

  // Zero the M-padding rows of WMMA A-operands once per launch.
  zero_k<<<PB, 256, 0, stream>>>(GA + GN, PN);
  zero_k<<<PB, 256, 0, stream>>>(GB + GN, PN);
  zero_k<<<PB, 256, 0, stream>>>(AGG + GN, PN);

  // ---- CNN branch ----
  conv5x5s2<<<(119072 + 255) / 256, 256, 0, stream>>>(x, conv1_w, conv1_b, c1,
                                                      1, 124, 124, 32, 61, 61, 1, 1);
  conv5x5s2<<<(26912 + 255) / 256, 256, 0, stream>>>(c1, conv2_w, conv2_b, c2,
                                                     32, 61, 61, 32, 29, 29, 0, 1);
  conv5x5s2<<<(6272 + 255) / 256, 256, 0, stream>>>(c2, conv3_w, conv3_b, c3,
                                                    32, 29, 29, 32, 14, 14, 1, 0);
  gemv_kernel<<<(1024 + 7) / 8, 256, 0, stream>>>(fc0_w, c3, fc0_b, fc0o, 1024, 6272, 1);
  gemv_kernel<<<(256 + 7) / 8, 256, 0, stream>>>(fc1_w, fc0o, fc1_b, f, 256, 1024, 1);

  // ---- state branch ----
  gemv_kernel<<<(64 + 7) / 8, 256, 0, stream>>>(fc2_w, state, fc2_b, s0, 64, 4, 1);
  gemv_kernel<<<(64 + 7) / 8, 256, 0, stream>>>(fc3_w, s0, fc3_b, f + 256, 64, 64, 1);

  // ---- graph branch ----
  s1l1_kernel<<<EB, 256, 0, stream>>>(xg, s1l1_wl, s1l1_bl, s1l1_wr, GA);

  // sage2: GB = relu(agg(GA)@wl^T + bl + GA@wr^T)
  sage_agg<<<EB, 256, 0, stream>>>(GA, AGG);
  set_rows_bias<<<EB, 256, 0, stream>>>(GB, s1l2_bl, GN);
  gemm19_wmma<<<GG, 32, 0, stream>>>(AGG, s1l2_wl, GB, GDIM, GDIM);
  gemm19_wmma<<<GG, 32, 0, stream>>>(GA, s1l2_wr, GB, GDIM, GDIM);
  relu_k<<<EB, 256, 0, stream>>>(GB, GN);

  // gat2: GA = relu(gat(GB))
  zero_k<<<EB, 256, 0, stream>>>(H, GN);
  gemm19_wmma<<<GG, 32, 0, stream>>>(GB, gat2_w, H, GDIM, GDIM);
  gat_scores<<<NNODE, 64, 0, stream>>>(H, gat2_as, gat2_ad, hs, hd);
  gat_softmax<<<1, 64, 0, stream>>>(hs, hd, alp);
  gat_agg<<<EB, 256, 0, stream>>>(H, alp, gat2_b, GA);

  // sage3: GB = relu(...)
  sage_agg<<<EB, 256, 0, stream>>>(GA, AGG);
  set_rows_bias<<<EB, 256, 0, stream>>>(GB, s3l1_bl, GN);
  gemm19_wmma<<<GG, 32, 0, stream>>>(AGG, s3l1_wl, GB, GDIM, GDIM);
  gemm19_wmma<<<GG, 32, 0, stream>>>(GA, s3l1_wr, GB, GDIM, GDIM);
  relu_k<<<EB, 256, 0, stream>>>(GB, GN);

  // sage4: GA = relu(...)
  sage_agg<<<EB, 256, 0, stream>>>(GB, AGG);
  set_rows_bias<<<EB, 256, 0, stream>>>(GA, s3l2_bl, GN);
  gemm19_wmma<<<GG, 32, 0, stream>>>(AGG, s3l2_wl, GA, GDIM, GDIM);
  gemm19_wmma<<<GG, 32, 0, stream>>>(GB, s3l2_wr, GA, GDIM, GDIM);
  relu_k<<<EB, 256, 0, stream>>>(GA, GN);

  // gat4: GB = relu(gat(GA))
  zero_k<<<EB, 256, 0, stream>>>(H, GN);
  gemm19_wmma<<<GG, 32, 0, stream>>>(GA, gat4_w, H, GDIM, GDIM);
  gat_scores<<<NNODE, 64, 0, stream>>>(H, gat4_as, gat4_ad, hs, hd);
  gat_softmax<<<1, 64, 0, stream>>>(hs, hd, alp);
  gat_agg<<<EB, 256, 0, stream>>>(H, alp, gat4_b, GB);

  // gcn5 + mean + fc_graph -> f[320:420]
  zero_k<<<EB, 256, 0, stream>>>(H, GN);
  gemm19_wmma<<<GG, 32, 0, stream>>>(GB, gcn5_w, H, GDIM, GDIM);
  gcn_agg<<<EB, 256, 0, stream>>>(H, gcn5_b, GA);
  mean19<<<(GDIM + 255) / 256, 256, 0, stream>>>(GA, gm);
  gemv_kernel<<<(100 + 7) / 8, 256, 0, stream>>>(fc_graph_w, gm, fc_graph_b, f + 320,
                                                 100, GDIM, 1);

  // ---- fusion + heads ----
  gemv_kernel<<<(256 + 7) / 8, 256, 0, stream>>>(fc_fusion_w, f, fc_fusion_b, fo,
                                                 256, 420, 1);
  heads_kernel<<<1, 64, 0, stream>>>(fo, fc_value_w, fc_value_b, fc_adv_w, fc_adv_b,
                                     (float*)d_out);
}